// GATv2Block_3805341024428
// MI455X (gfx1250) — compile-verified
//
#include <hip/hip_runtime.h>
#include <hip/hip_bf16.h>

#define F_IN   128
#define HC     128          // HEADS * C_OUT
#define HEADS  2
#define C_OUT  64
#define NEG_SLOPE 0.2f
#define EPS_GN 1e-5f

typedef __attribute__((ext_vector_type(2))) float v2f;
typedef __attribute__((ext_vector_type(8))) float v8f;

// ---------------- order-preserving float <-> uint for atomic max ----------------
__device__ __forceinline__ unsigned f2ord(float f) {
    unsigned u = __float_as_uint(f);
    return (u & 0x80000000u) ? ~u : (u | 0x80000000u);
}
__device__ __forceinline__ float ord2f(unsigned u) {
    return (u & 0x80000000u) ? __uint_as_float(u ^ 0x80000000u) : __uint_as_float(~u);
}

// ---------------- GEMM: C[n x 128] = A[n x 128] * B[128 x 128], f32 WMMA ----------------
// One wave computes one 16x16 output tile, K accumulated in steps of 4 via
// V_WMMA_F32_16X16X4_F32 (exact fp32 tensor path on CDNA5).
__global__ void gemm_f32_wmma_kernel(const float* __restrict__ A,
                                     const float* __restrict__ B,
                                     float* __restrict__ C,
                                     int n_rows) {
    const int waves_per_block = blockDim.x >> 5;
    int wave   = blockIdx.x * waves_per_block + (threadIdx.x >> 5);
    const int tiles_n = HC / 16;                 // 8 column tiles
    int tile_m = wave / tiles_n;
    int tile_n = wave % tiles_n;
    if (tile_m * 16 >= n_rows) return;           // wave-uniform guard (EXEC stays full)

    int lane = threadIdx.x & 31;
    int half = lane >> 4;                        // 0: K pair {0,1}; 1: K pair {2,3}
    int lm   = lane & 15;

    const float* arow = A + (size_t)(tile_m * 16 + lm) * F_IN;   // A row for this lane
    const float* bcol = B + (tile_n * 16 + lm);                  // B column for this lane

    v8f acc = {};
    #pragma unroll 4
    for (int k0 = 0; k0 < F_IN; k0 += 4) {
        int ka = k0 + half * 2;
        v2f a, b;
        a.x = arow[ka];
        a.y = arow[ka + 1];
        b.x = bcol[(size_t)ka * HC];
        b.y = bcol[(size_t)(ka + 1) * HC];
        acc = __builtin_amdgcn_wmma_f32_16x16x4_f32(
            /*neg_a=*/false, a, /*neg_b=*/false, b,
            /*c_mod=*/(short)0, acc, /*reuse_a=*/false, /*reuse_b=*/false);
    }
    // C/D layout: VGPR r -> lanes 0-15: M=r, N=lane ; lanes 16-31: M=r+8, N=lane-16
    int row0 = tile_m * 16 + half * 8;
    int col  = tile_n * 16 + lm;
    #pragma unroll
    for (int r = 0; r < 8; ++r)
        C[(size_t)(row0 + r) * HC + col] = acc[r];
}

// ---------------- per-edge attention logits + segment max ----------------
// One wave per edge. Lane covers channels {lane, lane+32} (head 0) and
// {lane+64, lane+96} (head 1). Wave-reduce, lane 0 stores + atomic-max.
__global__ void edge_logits_kernel(const float* __restrict__ xl,
                                   const float* __restrict__ xr,
                                   const int* __restrict__ ei,   // [2*E] : src rows then dst rows
                                   const float* __restrict__ att,// [HC]
                                   float* __restrict__ logits,   // [EP*HEADS]
                                   unsigned* __restrict__ segmax,// [N*HEADS] ordered-uint
                                   long long E, long long EP) {
    long long wid = (long long)blockIdx.x * (blockDim.x >> 5) + (threadIdx.x >> 5);
    if (wid >= EP) return;                       // wave-uniform
    int lane = threadIdx.x & 31;

    int s, d;
    if (wid < E) { s = ei[wid]; d = ei[E + wid]; }
    else         { s = d = (int)(wid - E); }     // self loop

    const float* pl = xl + (size_t)s * HC;
    const float* pr = xr + (size_t)d * HC;

    float s0 = 0.f, s1 = 0.f;
    #pragma unroll
    for (int j = 0; j < 4; ++j) {
        int t = lane + 32 * j;
        float v = pl[t] + pr[t];
        v = (v > 0.f) ? v : NEG_SLOPE * v;
        float contrib = v * att[t];
        if (j < 2) s0 += contrib; else s1 += contrib;
    }
    #pragma unroll
    for (int o = 16; o > 0; o >>= 1) {
        s0 += __shfl_xor(s0, o, 32);
        s1 += __shfl_xor(s1, o, 32);
    }
    if (lane == 0) {
        logits[wid * 2 + 0] = s0;
        logits[wid * 2 + 1] = s1;
        atomicMax(&segmax[(size_t)d * 2 + 0], f2ord(s0));
        atomicMax(&segmax[(size_t)d * 2 + 1], f2ord(s1));
    }
}

// ---------------- exp(logit - segmax) and segment sum ----------------
__global__ void edge_exp_kernel(const int* __restrict__ ei,
                                float* __restrict__ logits,       // in: logit, out: exp
                                const unsigned* __restrict__ segmax,
                                float* __restrict__ segsum,       // [N*HEADS]
                                long long E, long long EP) {
    long long i = (long long)blockIdx.x * blockDim.x + threadIdx.x;
    if (i >= EP * 2) return;
    long long e = i >> 1;
    int h = (int)(i & 1);
    int d = (e < E) ? ei[E + e] : (int)(e - E);
    float m  = ord2f(segmax[(size_t)d * 2 + h]);
    float ex = expf(logits[i] - m);
    logits[i] = ex;
    atomicAdd(&segsum[(size_t)d * 2 + h], ex);
}

// ---------------- scatter messages: out[dst] += alpha * xl[src] ----------------
// One 128-thread block per edge; thread t handles channel t (head = t>>6).
__global__ void edge_scatter_kernel(const float* __restrict__ xl,
                                    const int* __restrict__ ei,
                                    const float* __restrict__ expv,
                                    const float* __restrict__ segsum,
                                    float* __restrict__ out,      // [N*HC] accumulator
                                    long long E, long long EP) {
    long long e = (long long)blockIdx.x;
    if (e >= EP) return;
    int t = threadIdx.x;                 // 0..127
    int h = t >> 6;
    int s, d;
    if (e < E) { s = ei[e]; d = ei[E + e]; }
    else       { s = d = (int)(e - E); }
    float alpha = expv[e * 2 + h] / segsum[(size_t)d * 2 + h];
    atomicAdd(&out[(size_t)d * HC + t], alpha * xl[(size_t)s * HC + t]);
}

// ---------------- GraphNorm stats: per-channel mean & inv-std (one block / channel) ----
__global__ void norm_stats_kernel(const float* __restrict__ out,   // aggregated (no bias yet)
                                  const float* __restrict__ bias,
                                  const float* __restrict__ gms,   // gn_mean_scale
                                  float* __restrict__ msc,         // [HC] mean*scale
                                  float* __restrict__ invstd,      // [HC]
                                  int n) {
    int c   = blockIdx.x;
    int tid = threadIdx.x;
    float b = bias[c];
    float sum = 0.f, sq = 0.f;
    for (int i = tid; i < n; i += blockDim.x) {
        float v = out[(size_t)i * HC + c] + b;
        sum += v;
        sq  += v * v;
    }
    __shared__ float ss[256];
    __shared__ float sqs[256];
    ss[tid] = sum; sqs[tid] = sq;
    __syncthreads();
    for (int st = 128; st > 0; st >>= 1) {
        if (tid < st) { ss[tid] += ss[tid + st]; sqs[tid] += sqs[tid + st]; }
        __syncthreads();
    }
    if (tid == 0) {
        float m   = ss[0] / (float)n;
        float ev2 = sqs[0] / (float)n;
        float ms  = m * gms[c];
        float var = ev2 - 2.f * ms * m + ms * ms;   // E[(v - ms)^2]
        msc[c]    = ms;
        invstd[c] = rsqrtf(var + EPS_GN);
    }
}

// ---------------- final elementwise normalize (in place on d_out) ----------------
__global__ void finalize_kernel(float* __restrict__ out,
                                const float* __restrict__ bias,
                                const float* __restrict__ msc,
                                const float* __restrict__ invstd,
                                const float* __restrict__ gw,
                                const float* __restrict__ gb,
                                long long total) {
    long long i = (long long)blockIdx.x * blockDim.x + threadIdx.x;
    if (i >= total) return;
    int c = (int)(i & (HC - 1));
    float v = out[i] + bias[c];
    out[i] = gw[c] * (v - msc[c]) * invstd[c] + gb[c];
}

extern "C" void kernel_launch(void* const* d_in, const int* in_sizes, int n_in,
                              void* d_out, int out_size, void* d_ws, size_t ws_size,
                              hipStream_t stream) {
    const float* x     = (const float*)d_in[0];
    const int*   ei    = (const int*)  d_in[1];   // [2, E] flattened: src rows then dst rows
    const float* W_l   = (const float*)d_in[2];
    const float* W_r   = (const float*)d_in[3];
    const float* att   = (const float*)d_in[4];
    const float* bias  = (const float*)d_in[5];
    const float* gn_w  = (const float*)d_in[6];
    const float* gn_b  = (const float*)d_in[7];
    const float* gn_ms = (const float*)d_in[8];

    const int       n  = in_sizes[0] / F_IN;      // 50000
    const long long E  = in_sizes[1] / 2;         // 1600000
    const long long EP = E + n;                   // edges + self loops

    float* out = (float*)d_out;                   // [n, HC] accumulator then final

    // -------- workspace carve-up --------
    float*    xl     = (float*)d_ws;                         // n*HC
    float*    xr     = xl + (size_t)n * HC;                  // n*HC
    float*    logits = xr + (size_t)n * HC;                  // EP*HEADS
    unsigned* segmax = (unsigned*)(logits + (size_t)EP * HEADS); // n*HEADS
    float*    segsum = (float*)(segmax + (size_t)n * HEADS); // n*HEADS
    float*    msc    = segsum + (size_t)n * HEADS;           // HC
    float*    invstd = msc + HC;                             // HC

    // -------- init (graph-capture safe) --------
    hipMemsetAsync(out, 0, (size_t)n * HC * sizeof(float), stream);
    // segmax ordered-uint 0 == "below every float"; segsum 0.0f — contiguous region
    hipMemsetAsync(segmax, 0, (size_t)n * HEADS * 2 * sizeof(float), stream);

    // -------- dense projections via fp32 WMMA --------
    {
        int tiles   = ((n + 15) / 16) * (HC / 16);   // 16x16 tiles, one wave each
        int wpb     = 8;                             // 256 threads = 8 waves
        int blocks  = (tiles + wpb - 1) / wpb;
        gemm_f32_wmma_kernel<<<blocks, 32 * wpb, 0, stream>>>(x, W_l, xl, n);
        gemm_f32_wmma_kernel<<<blocks, 32 * wpb, 0, stream>>>(x, W_r, xr, n);
    }

    // -------- edge phase --------
    {
        int wpb    = 8;                              // 8 waves per 256-thread block
        long long blocks = (EP + wpb - 1) / wpb;
        edge_logits_kernel<<<(unsigned)blocks, 32 * wpb, 0, stream>>>(
            xl, xr, ei, att, logits, segmax, E, EP);
    }
    {
        long long threads = EP * 2;
        long long blocks  = (threads + 255) / 256;
        edge_exp_kernel<<<(unsigned)blocks, 256, 0, stream>>>(
            ei, logits, segmax, segsum, E, EP);
    }
    edge_scatter_kernel<<<(unsigned)EP, HC, 0, stream>>>(
        xl, ei, logits, segsum, out, E, EP);

    // -------- GraphNorm --------
    norm_stats_kernel<<<HC, 256, 0, stream>>>(out, bias, gn_ms, msc, invstd, n);
    {
        long long total  = (long long)n * HC;
        long long blocks = (total + 255) / 256;
        finalize_kernel<<<(unsigned)blocks, 256, 0, stream>>>(
            out, bias, msc, invstd, gn_w, gn_b, total);
    }
}